// UnsupervisedContextNormalization_29437705847527
// MI455X (gfx1250) — compile-verified
//
#include <hip/hip_runtime.h>
#include <hip/hip_bf16.h>

// Problem constants (from reference): x (32,128,128,64) f32, K=8, EPS=1e-3
#define BB   32
#define HH   128
#define WW   128
#define CC   64
#define KK   8
#define HWP  (HH * WW)          // 16384 positions per batch
#define EPSF 1e-3f
#define LOG2E 1.4426950408889634f

typedef float v2f __attribute__((ext_vector_type(2)));
typedef float v8f __attribute__((ext_vector_type(8)));

// ---------------------------------------------------------------------------
// Kernel 0: per (k,c) quadratic coefficients for the exp2-form Gaussian:
//   u = exp2(a2*x^2 + a1*x + a0) = exp(-0.5 (x-m)^2 / (softplus(var)+eps))
//   a2 = log2e * (-0.5/(sp+eps));  a1 = -2*a2*m;  a0 = a2*m^2
// ---------------------------------------------------------------------------
__global__ void ucn_prep_kernel(const float* __restrict__ variance,
                                const float* __restrict__ mean,
                                float* __restrict__ A2, float* __restrict__ A1,
                                float* __restrict__ A0) {
    int j = blockIdx.x * blockDim.x + threadIdx.x;
    if (j < KK * CC) {
        float v  = variance[j];
        float sp = logf(1.0f + __expf(v));      // softplus
        float cf = -0.5f / (sp + EPSF);
        float m  = mean[j];
        float a2 = cf * LOG2E;
        A2[j] = a2;
        A1[j] = -2.0f * a2 * m;
        A0[j] = a2 * m * m;
    }
}

// ---------------------------------------------------------------------------
// Kernel 1: fused statistics pass over x.
//   Per (k,c):  S   = sum tau          (global over b,h,w)
//               E'  = sum tau * x      (per batch)
//               V'  = sum tau^3 * x^2  (per batch)
// Grid: 32 batches * 16 chunks = 512 blocks, 256 threads.
// Each block covers 1024 (h,w) positions of one batch; tid%64 -> channel.
// Regular-temporal x loads (keep x resident in the 192MB L2 for pass 3).
// Deterministic LDS tree reduction; per-block partials to workspace.
// ---------------------------------------------------------------------------
__global__ __launch_bounds__(256) void ucn_stats_kernel(
    const float* __restrict__ x, const float* __restrict__ A2,
    const float* __restrict__ A1, const float* __restrict__ A0,
    float* __restrict__ partS, float* __restrict__ partE,
    float* __restrict__ partV) {
    const int tid   = threadIdx.x;
    const int c     = tid & 63;
    const int g     = tid >> 6;          // 0..3
    const int blk   = blockIdx.x;
    const int b     = blk >> 4;
    const int chunk = blk & 15;

    float a2[KK], a1[KK], a0[KK];
#pragma unroll
    for (int k = 0; k < KK; ++k) {
        a2[k] = A2[k * CC + c];
        a1[k] = A1[k * CC + c];
        a0[k] = A0[k * CC + c];
    }

    float acc[3][KK];
#pragma unroll
    for (int s = 0; s < 3; ++s)
#pragma unroll
        for (int k = 0; k < KK; ++k) acc[s][k] = 0.0f;

    const int pbase = b * HWP + chunk * 1024;
    const float* xp = x + (size_t)pbase * CC + c;

    for (int i = 0; i < 256; ++i) {
        int p = g + 4 * i;                       // position within chunk
        float xv = xp[(size_t)p * CC];
        float u[KK];
        float den = EPSF;                        // seed with eps: den+eps free
#pragma unroll
        for (int k = 0; k < KK; ++k) {
            float arg = fmaf(xv, fmaf(a2[k], xv, a1[k]), a0[k]);
            u[k] = __builtin_amdgcn_exp2f(arg);  // v_exp_f32, no scale mul
            den += u[k];
        }
        float inv = __builtin_amdgcn_rcpf(den);
#pragma unroll
        for (int k = 0; k < KK; ++k) {
            float tau = u[k] * inv;
            float te  = tau * xv;
            acc[0][k] += tau;                       // S
            acc[1][k] += te;                        // sum tau*x
            acc[2][k]  = fmaf(te * te, tau, acc[2][k]);  // sum tau^3*x^2
        }
    }

    __shared__ float red[4 * KK * CC];           // 2048 floats
    float* outs[3] = {partS, partE, partV};
    for (int s = 0; s < 3; ++s) {
        __syncthreads();
#pragma unroll
        for (int k = 0; k < KK; ++k) red[g * 512 + k * CC + c] = acc[s][k];
        __syncthreads();
        for (int j = tid; j < 512; j += 256) {
            float v = red[j] + red[512 + j] + red[1024 + j] + red[1536 + j];
            outs[s][(size_t)blk * 512 + j] = v;
        }
    }
}

// ---------------------------------------------------------------------------
// Kernel 2a: S[j] = sum over 512 block-partials of partS[blk][j], j in [0,512)
// V_WMMA_F32_16X16X4_F32: A = ones(16x4), B = 4 partial rows x 16 cols.
// D[m][n] accumulates the column sums (identical in every row m), invariant
// to the exact B row->VGPR-slot bijection.  One wave per 16 columns.
// ---------------------------------------------------------------------------
__global__ __launch_bounds__(32) void ucn_colsum_wmma_kernel(
    const float* __restrict__ part, float* __restrict__ S) {
    const int lane = threadIdx.x;                // 0..31, full wave32
    const int col  = blockIdx.x * 16 + (lane & 15);
    const int half = lane >> 4;

    v8f acc = {};
    v2f a;
    a.x = 1.0f;
    a.y = 1.0f;                                   // A = all ones
    for (int r = 0; r < 512; r += 4) {
        int r0 = r + 2 * half;                    // this half supplies 2 rows
        v2f bb;
        bb.x = part[(size_t)r0 * 512 + col];
        bb.y = part[(size_t)(r0 + 1) * 512 + col];
        acc = __builtin_amdgcn_wmma_f32_16x16x4_f32(
            /*neg_a=*/false, a, /*neg_b=*/false, bb,
            /*c_mod=*/(short)0, acc, /*reuse_a=*/false, /*reuse_b=*/false);
    }
    // D VGPR0: lanes 0-15 hold M=0,N=lane  -> the column sum.
    if (lane < 16) S[col] = acc[0];
}

// ---------------------------------------------------------------------------
// Kernel 2b: fold 1/(S+eps) powers into per-(b,k,c) tables.
//   E[b,k,c]  = (sum tau*x)      / (S+eps)   / 16384
//   V[b,k,c]  = (sum tau^3*x^2)  / (S+eps)^3 / 16384
//   tabA      = c0 * rsqrt(V + eps),  c0 = rsqrt(1+eps)  (softmax(prior)==1)
// 16384 threads.
// ---------------------------------------------------------------------------
__global__ __launch_bounds__(256) void ucn_tables_kernel(
    const float* __restrict__ partE, const float* __restrict__ partV,
    const float* __restrict__ S, float* __restrict__ tabE,
    float* __restrict__ tabA) {
    int idx = blockIdx.x * 256 + threadIdx.x;    // < 32*512
    int b = idx >> 9;
    int j = idx & 511;
    float se = 0.0f, sv = 0.0f;
    for (int ch = 0; ch < 16; ++ch) {
        size_t o = (size_t)((b << 4) + ch) * 512 + j;
        se += partE[o];
        sv += partV[o];
    }
    float r = 1.0f / (S[j] + EPSF);
    float E = se * r * (1.0f / (float)HWP);
    float V = sv * (r * r * r) * (1.0f / (float)HWP);
    const float c0 = rsqrtf(1.0f + EPSF);
    tabE[idx] = E;
    tabA[idx] = c0 * rsqrtf(V + EPSF);
}

// ---------------------------------------------------------------------------
// Kernel 3: output pass.
//   out = (sum_k u_k * (x*A[b,k,c] - E[b,k,c]*A[b,k,c])) / (sum_k u_k + eps)
// Per k: 2 FMA (exp arg) + exp2 + add + 2 FMA = 5 VALU + 1 TRANS.
// NT loads for x (last use) and NT stores for out.
// 2048 blocks x 256 threads, 256 positions per block.
// ---------------------------------------------------------------------------
__global__ __launch_bounds__(256) void ucn_out_kernel(
    const float* __restrict__ x, const float* __restrict__ A2,
    const float* __restrict__ A1, const float* __restrict__ A0,
    const float* __restrict__ tabE, const float* __restrict__ tabA,
    float* __restrict__ out) {
    const int tid = threadIdx.x;
    const int c   = tid & 63;
    const int g   = tid >> 6;
    const int p0  = blockIdx.x * 256;            // global position base
    const int b   = p0 >> 14;                    // 16384 positions per batch

    float a2[KK], a1[KK], a0[KK], A[KK], nEA[KK];
#pragma unroll
    for (int k = 0; k < KK; ++k) {
        a2[k] = A2[k * CC + c];
        a1[k] = A1[k * CC + c];
        a0[k] = A0[k * CC + c];
        float Av = tabA[(b * KK + k) * CC + c];
        float Ev = tabE[(b * KK + k) * CC + c];
        A[k]   = Av;
        nEA[k] = -Ev * Av;
    }

    const float* xp = x + (size_t)p0 * CC + c;
    float* op       = out + (size_t)p0 * CC + c;

    for (int i = 0; i < 64; ++i) {
        size_t off = (size_t)(g + 4 * i) * CC;
        float xv = __builtin_nontemporal_load(xp + off);
        float den = EPSF, num = 0.0f;
#pragma unroll
        for (int k = 0; k < KK; ++k) {
            float arg = fmaf(xv, fmaf(a2[k], xv, a1[k]), a0[k]);
            float u   = __builtin_amdgcn_exp2f(arg);
            den += u;
            float t = fmaf(xv, A[k], nEA[k]);    // A*(x - E)
            num = fmaf(u, t, num);
        }
        float r = __builtin_amdgcn_rcpf(den);
        __builtin_nontemporal_store(num * r, op + off);
    }
}

// ---------------------------------------------------------------------------
// Launch: prep -> stats -> WMMA colsum -> tables -> out   (all on `stream`)
// ---------------------------------------------------------------------------
extern "C" void kernel_launch(void* const* d_in, const int* in_sizes, int n_in,
                              void* d_out, int out_size, void* d_ws,
                              size_t ws_size, hipStream_t stream) {
    const float* x        = (const float*)d_in[0];
    const float* mean     = (const float*)d_in[1];
    const float* variance = (const float*)d_in[2];
    // d_in[3] (prior) is unused: softmax over a length-1 axis is exactly 1.0.
    float* out = (float*)d_out;

    float* ws    = (float*)d_ws;
    float* A2    = ws;                      // 512
    float* A1    = A2 + 512;                // 512
    float* A0    = A1 + 512;                // 512
    float* partS = A0 + 512;                // 512 blocks * 512
    float* partE = partS + 512 * 512;       // 512 * 512
    float* partV = partE + 512 * 512;       // 512 * 512
    float* Sbuf  = partV + 512 * 512;       // 512
    float* tabE  = Sbuf + 512;              // 32*8*64 = 16384
    float* tabA  = tabE + BB * KK * CC;     // 16384
    // total: ~821K floats (~3.2 MiB) of workspace

    ucn_prep_kernel<<<2, 256, 0, stream>>>(variance, mean, A2, A1, A0);
    ucn_stats_kernel<<<BB * 16, 256, 0, stream>>>(x, A2, A1, A0, partS, partE,
                                                  partV);
    ucn_colsum_wmma_kernel<<<32, 32, 0, stream>>>(partS, Sbuf);
    ucn_tables_kernel<<<64, 256, 0, stream>>>(partE, partV, Sbuf, tabE, tabA);
    ucn_out_kernel<<<2048, 256, 0, stream>>>(x, A2, A1, A0, tabE, tabA, out);
}